// LocalFeatureAggregation_34943853920370
// MI455X (gfx1250) — compile-verified
//
#include <hip/hip_runtime.h>
#include <hip/hip_bf16.h>
#include <math.h>

typedef float v2f __attribute__((ext_vector_type(2)));
typedef float v4f __attribute__((ext_vector_type(4)));
typedef float v8f __attribute__((ext_vector_type(8)));

#define BATCH 4
#define NPTS  8192
#define KNN_K 16
#define CIN   64
#define CMID  64
#define COUT  128
#define ROWS  (BATCH * NPTS)   // 32768

// ---------------------------------------------------------------------------
// Kernel 1: KNN via V_WMMA_F32_16X16X4_F32
// Block = 256 threads = 8 waves; each wave owns 16 queries.
// A (16x4) = candidate coords {x,y,z,0}, B (4x16) = query coords.
// C layout: lane l -> query n = l&15; VGPR r -> candidate row r (+8 upper half).
// Rank key: |c|^2 - 2 c.q (per-query |q|^2 is rank-invariant, dropped).
// Candidates packed in LDS as float4 {x,y,z,0} -> A-frag is ONE ds_load_b64,
// no divergence. Top-16 kept per lane with a branchless compare-select
// insertion, gated by (s < current worst) so it rarely executes.
// ---------------------------------------------------------------------------
__global__ __launch_bounds__(256) void knn_kernel(const float* __restrict__ coords,
                                                  int* __restrict__ knn_idx) {
    __shared__ float cpak[256][4];          // {x, y, z, 0} per candidate
    __shared__ float lsq[256];              // |c|^2 per candidate
    __shared__ float mrg_d[8][32][KNN_K];
    __shared__ int   mrg_i[8][32][KNN_K];

    const int tid  = threadIdx.x;
    const int wave = tid >> 5;
    const int lane = tid & 31;
    const int half = lane >> 4;             // 0: K=0,1 rows 0..7 | 1: K=2,3 rows 8..15
    const int qn   = lane & 15;

    const int tilesPerB = NPTS / 128;
    const int b     = blockIdx.x / tilesPerB;
    const int qbase = (blockIdx.x % tilesPerB) * 128 + wave * 16;

    // B fragment: query coords (constant across the candidate loop)
    const size_t qoff = ((size_t)b * NPTS + qbase + qn) * 3;
    const float qx = coords[qoff + 0], qy = coords[qoff + 1], qz = coords[qoff + 2];
    v2f bfrag;
    bfrag.x = half ? qz : qx;               // K=0 (x) / K=2 (z)
    bfrag.y = half ? 0.0f : qy;             // K=1 (y) / K=3 (pad)

    float bd[KNN_K];
    int   bi[KNN_K];
#pragma unroll
    for (int j = 0; j < KNN_K; ++j) { bd[j] = 3.0e38f; bi[j] = 0; }

    for (int cbase = 0; cbase < NPTS; cbase += 256) {
        {
            const size_t coff = ((size_t)b * NPTS + cbase + tid) * 3;
            const float cx = coords[coff + 0], cy = coords[coff + 1], cz = coords[coff + 2];
            cpak[tid][0] = cx; cpak[tid][1] = cy; cpak[tid][2] = cz; cpak[tid][3] = 0.0f;
            lsq[tid] = cx * cx + cy * cy + cz * cz;
            if (cbase + 256 < NPTS) {
                // prefetch next candidate chunk (global_prefetch_b8)
                __builtin_prefetch(&coords[coff + 768], 0, 3);
            }
        }
        __syncthreads();

        for (int blk = 0; blk < 16; ++blk) {
            const int mb = blk * 16;
            // A fragment: one branchless 8-byte LDS load ({x,y} or {z,0})
            const v2f af = *(const v2f*)(&cpak[mb + qn][half * 2]);
            // candidate squared norms for this lane's 8 rows (two b128 loads)
            const v4f q0 = *(const v4f*)(&lsq[mb + half * 8]);
            const v4f q1 = *(const v4f*)(&lsq[mb + half * 8 + 4]);

            v8f c = {0.f, 0.f, 0.f, 0.f, 0.f, 0.f, 0.f, 0.f};
            c = __builtin_amdgcn_wmma_f32_16x16x4_f32(false, af, false, bfrag,
                                                      (short)0, c, false, false);
#pragma unroll
            for (int r = 0; r < 8; ++r) {
                const float nrm = (r < 4) ? q0[r & 3] : q1[r & 3];
                const float s = fmaf(-2.0f, c[r], nrm);
                if (s < bd[KNN_K - 1]) {                    // rare path
                    float cd = s;
                    int   ci = cbase + mb + half * 8 + r;
#pragma unroll
                    for (int j = 0; j < KNN_K; ++j) {       // branchless v_cndmask chain
                        const bool sw = cd < bd[j];
                        const float nd = sw ? cd : bd[j];
                        const int   ni = sw ? ci : bi[j];
                        cd = sw ? bd[j] : cd;
                        ci = sw ? bi[j] : ci;
                        bd[j] = nd; bi[j] = ni;
                    }
                }
            }
        }
        __syncthreads();
    }

    // Merge the two half-wave candidate streams for each query via LDS.
#pragma unroll
    for (int j = 0; j < KNN_K; ++j) { mrg_d[wave][lane][j] = bd[j]; mrg_i[wave][lane][j] = bi[j]; }
    __syncthreads();

    if (lane < 16) {
#pragma unroll
        for (int j = 0; j < KNN_K; ++j) {
            float cd = mrg_d[wave][lane + 16][j];
            int   ci = mrg_i[wave][lane + 16][j];
            if (cd < bd[KNN_K - 1]) {
#pragma unroll
                for (int jj = 0; jj < KNN_K; ++jj) {
                    const bool sw = cd < bd[jj];
                    const float nd = sw ? cd : bd[jj];
                    const int   ni = sw ? ci : bi[jj];
                    cd = sw ? bd[jj] : cd;
                    ci = sw ? bi[jj] : ci;
                    bd[jj] = nd; bi[jj] = ni;
                }
            }
        }
        const size_t o = ((size_t)b * NPTS + qbase + lane) * KNN_K;
#pragma unroll
        for (int j = 0; j < KNN_K; ++j) knn_idx[o + j] = bi[j];
    }
}

// ---------------------------------------------------------------------------
// Kernel 2: gather 16 neighbor feature rows + mean -> X (ROWS x 64)
// one thread per (point, channel)
// ---------------------------------------------------------------------------
__global__ void gather_mean(const float* __restrict__ feats,
                            const int* __restrict__ knn,
                            float* __restrict__ X) {
    const int i = blockIdx.x * blockDim.x + threadIdx.x;
    if (i >= ROWS * CIN) return;
    const int ch = i & (CIN - 1);
    const int p  = i >> 6;
    const int b  = p / NPTS;
    const int* row = knn + (size_t)p * KNN_K;
    float acc = 0.0f;
#pragma unroll
    for (int j = 0; j < KNN_K; ++j) {
        const int nb = row[j];
        acc += feats[(((size_t)b * NPTS) + nb) * CIN + ch];
    }
    X[(size_t)p * CIN + ch] = acc * (1.0f / 16.0f);
}

// ---------------------------------------------------------------------------
// Kernel 3: GEMM (ROWS x 64) @ (64 x NCH) + bias via f32 WMMA,
// with per-channel sum / sumsq accumulation for BatchNorm.
// Block: 256 threads = 8 waves, 128 rows per block. W staged transposed in LDS
// so both A and B fragments are single 8-byte loads per k-step.
// ---------------------------------------------------------------------------
template <int NCH>
__global__ __launch_bounds__(256) void gemm_bn(const float* __restrict__ X,
                                               const float* __restrict__ W,
                                               const float* __restrict__ bias,
                                               float* __restrict__ Y,
                                               float* __restrict__ gsum,
                                               float* __restrict__ gsum2) {
    __shared__ float Wt[64 * NCH];          // Wt[c*64 + k]
    __shared__ float sL[NCH], s2L[NCH];

    const int tid  = threadIdx.x;
    const int wave = tid >> 5;
    const int lane = tid & 31;
    const int half = lane >> 4;
    const int ln   = lane & 15;

    for (int i = tid; i < 64 * NCH; i += 256) {
        const int k = i / NCH, c = i % NCH;
        Wt[c * 64 + k] = W[i];
    }
    if (tid < NCH) { sL[tid] = 0.0f; s2L[tid] = 0.0f; }
    __syncthreads();

    const int rowA = blockIdx.x * 128 + wave * 16 + ln;
    const float* xrow = X + (size_t)rowA * 64;
    const int off = half * 2;               // K split across wave halves

    v8f acc[NCH / 16];
#pragma unroll
    for (int ct = 0; ct < NCH / 16; ++ct)
        acc[ct] = (v8f){0.f, 0.f, 0.f, 0.f, 0.f, 0.f, 0.f, 0.f};

#pragma unroll
    for (int k = 0; k < 64; k += 4) {
        const v2f af = *(const v2f*)(xrow + k + off);
#pragma unroll
        for (int ct = 0; ct < NCH / 16; ++ct) {
            const int col = ct * 16 + ln;
            const v2f bf = *(const v2f*)(&Wt[col * 64 + k + off]);
            acc[ct] = __builtin_amdgcn_wmma_f32_16x16x4_f32(false, af, false, bf,
                                                            (short)0, acc[ct], false, false);
        }
    }

    const int rbase = blockIdx.x * 128 + wave * 16 + half * 8;
#pragma unroll
    for (int ct = 0; ct < NCH / 16; ++ct) {
        const int col = ct * 16 + ln;
        const float cb = bias[col];
        float ls = 0.0f, ls2 = 0.0f;
#pragma unroll
        for (int r = 0; r < 8; ++r) {
            const float y = acc[ct][r] + cb;
            Y[(size_t)(rbase + r) * NCH + col] = y;
            ls += y; ls2 += y * y;
        }
        atomicAdd(&sL[col], ls);
        atomicAdd(&s2L[col], ls2);
    }
    __syncthreads();
    if (tid < NCH) { atomicAdd(&gsum[tid], sL[tid]); atomicAdd(&gsum2[tid], s2L[tid]); }
}

// ---------------------------------------------------------------------------
// Small helpers: zero stats, finalize BN scale/shift, elementwise applies
// ---------------------------------------------------------------------------
__global__ void zero_f32(float* __restrict__ p, int n) {
    const int i = blockIdx.x * blockDim.x + threadIdx.x;
    if (i < n) p[i] = 0.0f;
}

__global__ void finalize_stats(const float* __restrict__ sum, const float* __restrict__ sumsq,
                               const float* __restrict__ gamma, const float* __restrict__ beta,
                               float* __restrict__ scl, float* __restrict__ shift, int nch) {
    const int c = threadIdx.x;
    if (c >= nch) return;
    const float inv = 1.0f / (float)ROWS;
    const float m = sum[c] * inv;
    const float v = sumsq[c] * inv - m * m;
    const float s = gamma[c] * rsqrtf(v + 1e-5f);
    scl[c] = s;
    shift[c] = beta[c] - m * s;
}

// out = relu(y*scl+shift)
__global__ void apply_relu(const float* __restrict__ Y, const float* __restrict__ scl,
                           const float* __restrict__ shift, float* __restrict__ out,
                           int nch, int total) {
    const int i = blockIdx.x * blockDim.x + threadIdx.x;
    if (i >= total) return;
    const int c = i & (nch - 1);
    const float v = Y[i] * scl[c] + shift[c];
    out[i] = v > 0.0f ? v : 0.0f;
}

// attn = sigmoid(relu(ya*scl+shift)); out = agg*attn + agg
__global__ void apply_attn(const float* __restrict__ Ya, const float* __restrict__ agg,
                           const float* __restrict__ scl, const float* __restrict__ shift,
                           float* __restrict__ out, int total) {
    const int i = blockIdx.x * blockDim.x + threadIdx.x;
    if (i >= total) return;
    const int c = i & (CMID - 1);
    float t = Ya[i] * scl[c] + shift[c];
    t = t > 0.0f ? t : 0.0f;
    const float a = 1.0f / (1.0f + __expf(-t));
    out[i] = agg[i] * (a + 1.0f);
}

// ---------------------------------------------------------------------------
extern "C" void kernel_launch(void* const* d_in, const int* in_sizes, int n_in,
                              void* d_out, int out_size, void* d_ws, size_t ws_size,
                              hipStream_t stream) {
    const float* coords = (const float*)d_in[0];
    const float* feats  = (const float*)d_in[1];
    // d_in[2] = k (==16, baked in)
    const float* W1  = (const float*)d_in[3];
    const float* b1  = (const float*)d_in[4];
    const float* g1  = (const float*)d_in[5];
    const float* be1 = (const float*)d_in[6];
    const float* Wa  = (const float*)d_in[7];
    const float* ba  = (const float*)d_in[8];
    const float* ga  = (const float*)d_in[9];
    const float* bea = (const float*)d_in[10];
    const float* W2  = (const float*)d_in[11];
    const float* b2  = (const float*)d_in[12];
    const float* g2  = (const float*)d_in[13];
    const float* be2 = (const float*)d_in[14];
    float* out = (float*)d_out;

    // workspace carve-up
    char* ws = (char*)d_ws;
    int*   idxBuf = (int*)ws;                 ws += (size_t)ROWS * KNN_K * sizeof(int);    // 2 MB
    float* bufA   = (float*)ws;               ws += (size_t)ROWS * CMID * sizeof(float);   // 8 MB  (X, later Ya)
    float* bufB   = (float*)ws;               ws += (size_t)ROWS * CMID * sizeof(float);   // 8 MB  (Y1, later fused)
    float* bufC   = (float*)ws;               ws += (size_t)ROWS * CMID * sizeof(float);   // 8 MB  (agg)
    float* stats  = (float*)ws;               // 3 stages x 512 floats
    float* sum0 = stats + 0 * 512, *sq0 = sum0 + 128, *scl0 = sum0 + 256, *sh0 = sum0 + 384;
    float* sum1 = stats + 1 * 512, *sq1 = sum1 + 128, *scl1 = sum1 + 256, *sh1 = sum1 + 384;
    float* sum2 = stats + 2 * 512, *sq2 = sum2 + 128, *scl2 = sum2 + 256, *sh2 = sum2 + 384;

    const int totMid = ROWS * CMID;   // 2M
    const int totOut = ROWS * COUT;   // 4M

    // 1) KNN (WMMA distance blocks)
    knn_kernel<<<ROWS / 128, 256, 0, stream>>>(coords, idxBuf);
    // 2) gather + mean -> X
    gather_mean<<<(ROWS * CIN + 255) / 256, 256, 0, stream>>>(feats, idxBuf, bufA);
    // 3) zero BN accumulators (every call: graph-replay safe)
    zero_f32<<<6, 256, 0, stream>>>(stats, 3 * 512);
    // 4) Y1 = X @ W1 + b1  (+ stats)
    gemm_bn<CMID><<<ROWS / 128, 256, 0, stream>>>(bufA, W1, b1, bufB, sum0, sq0);
    finalize_stats<<<1, CMID, 0, stream>>>(sum0, sq0, g1, be1, scl0, sh0, CMID);
    // 5) agg = relu(norm(Y1))
    apply_relu<<<(totMid + 255) / 256, 256, 0, stream>>>(bufB, scl0, sh0, bufC, CMID, totMid);
    // 6) Ya = agg @ Wa + ba (+ stats)
    gemm_bn<CMID><<<ROWS / 128, 256, 0, stream>>>(bufC, Wa, ba, bufA, sum1, sq1);
    finalize_stats<<<1, CMID, 0, stream>>>(sum1, sq1, ga, bea, scl1, sh1, CMID);
    // 7) fused = agg * sigmoid(relu(norm(Ya))) + agg
    apply_attn<<<(totMid + 255) / 256, 256, 0, stream>>>(bufA, bufC, scl1, sh1, bufB, totMid);
    // 8) Y2 = fused @ W2 + b2 -> d_out (+ stats)
    gemm_bn<COUT><<<ROWS / 128, 256, 0, stream>>>(bufB, W2, b2, out, sum2, sq2);
    finalize_stats<<<1, COUT, 0, stream>>>(sum2, sq2, g2, be2, scl2, sh2, COUT);
    // 9) out = relu(norm(Y2)) in-place
    apply_relu<<<(totOut + 255) / 256, 256, 0, stream>>>(out, scl2, sh2, out, COUT, totOut);
}